// o3_tensor_product_23837068492877
// MI455X (gfx1250) — compile-verified
//
#include <hip/hip_runtime.h>

typedef float v2f __attribute__((ext_vector_type(2)));
typedef float v8f __attribute__((ext_vector_type(8)));

#define NPATHS 44

// Path tables in (l1,l2,l3) loop order — matches _w3j_keys() / concat order.
#define L1_INIT {0,0,0,0, 1,1,1,1,1,1,1,1,1,1, 2,2,2,2,2,2,2,2,2,2,2,2,2,2, 3,3,3,3,3,3,3,3,3,3,3,3,3,3,3,3}
#define L2_INIT {0,1,2,3, 0,1,1,1,2,2,2,3,3,3, 0,1,1,1,2,2,2,2,2,3,3,3,3,3, 0,1,1,1,2,2,2,2,2,3,3,3,3,3,3,3}
#define L3_INIT {0,1,2,3, 1,0,1,2,1,2,3,2,3,4, 2,1,2,3,0,1,2,3,4,1,2,3,4,5, 3,2,3,4,1,2,3,4,5,0,1,2,3,4,5,6}
// Position of each path inside its l3 output-concat group (loop order).
#define PSLOT_INIT {0,0,0,0, 1,1,2,1,3,2,1,3,2,0, 4,4,5,3,2,5,6,4,1,6,7,5,2,0, 6,8,7,3,7,9,8,4,1,3,8,10,9,5,2,0}

__constant__ int c_l1[NPATHS]    = L1_INIT;
__constant__ int c_l2[NPATHS]    = L2_INIT;
__constant__ int c_l3[NPATHS]    = L3_INIT;
__constant__ int c_pslot[NPATHS] = PSLOT_INIT;
// paths per l3 group, and flat-float base offset of each l3 output tensor
__constant__ int c_P[7]    = {4, 9, 11, 10, 6, 3, 1};
__constant__ int c_base[7] = {0, 2097152, 16252928, 45088768, 81788928, 110100480, 127401984};

struct TPParams {
    const float* x1[4];
    const float* x2[4];
    const float* w3j[NPATHS];
    float* out;
};

// One workgroup per (path, batch). 4 waves of 32.
__global__ __launch_bounds__(128) void o3_tp_wmma_kernel(TPParams p) {
    const int path = blockIdx.x;
    const int b    = blockIdx.y;
    const int l1 = c_l1[path], l2 = c_l2[path], l3 = c_l3[path];
    const int D1 = 2 * l1 + 1, D2 = 2 * l2 + 1, D3 = 2 * l3 + 1;
    const int P = c_P[l3];
    const int pslot = c_pslot[path];
    const long long obase =
        (long long)c_base[l3] + (long long)(b * P + pslot) * 1024LL * (long long)D3;

    // LDS: z rows padded to K=8; row stride 464 (== 16 mod 64 banks -> rows 2
    // apart land 32 banks apart: lanes 0-15 vs 16-31 never conflict on B loads)
    __shared__ float zs[8 * 464];      // B operand, K-padded
    __shared__ float x1s[32 * 9];      // A operand, K-padded, stride 9 (conflict-free)
    __shared__ float w3s[640];         // w3j for this path (<= 637 floats)
    __shared__ float x2s[224];         // x2[b] rows (<= 32*7 floats)

    const int tid = threadIdx.x;
    const float* __restrict__ w3  = p.w3j[path];
    const float* __restrict__ x1p = p.x1[l1] + (size_t)b * 32 * D1;
    const float* __restrict__ x2p = p.x2[l2] + (size_t)b * 32 * D2;

    // ---- Stage A0: pull tiny operands into LDS ----
    for (int t = tid; t < D1 * D2 * D3; t += 128) w3s[t] = w3[t];
    for (int t = tid; t < 32 * D2; t += 128)      x2s[t] = x2p[t];
    for (int t = tid; t < 32 * 8; t += 128) {     // A tile, zero-pad K to 8
        int i = t >> 3, k = t & 7;
        x1s[i * 9 + k] = (k < D1) ? x1p[i * D1 + k] : 0.0f;
    }
    __syncthreads();

    // ---- Stage A1: z[m][col] = sum_n w3j[m][n][M(col)] * x2[j(col)][n] ----
    const int NC = 32 * D3;  // flattened (j,M) columns
    for (int t = tid; t < 8 * NC; t += 128) {
        int m = t / NC, col = t - m * NC;
        int j = col / D3, M = col - j * D3;
        float acc = 0.0f;
        if (m < D1) {
            const float* wrow = &w3s[(m * D2) * D3 + M];
            const float* xrow = &x2s[j * D2];
            for (int n = 0; n < D2; ++n) acc += wrow[n * D3] * xrow[n];
        }
        zs[m * 464 + col] = acc;
    }
    __syncthreads();

    // ---- Stage B: C[32, 32*D3] = A[32, K] x B[K, 32*D3] via f32 WMMA ----
    const int wave = tid >> 5;
    const int lane = tid & 31;
    const int row  = lane & 15;   // A row / B-C column within tile
    const int h    = lane >> 4;   // lane half
    const int nTiles = 4 * D3;    // 2 i-tiles x 2*D3 col-tiles
    for (int t = wave; t < nTiles; t += 4) {
        const int it = t & 1;
        const int nt = t >> 1;
        const int colg = nt * 16 + row;          // global column (j*D3 + M)
        const int j = colg / D3;
        const int M = colg - j * D3;

        // A fragment (ISA 7.12.2: lane half h holds K = 2h, 2h+1)
        const float* arow = &x1s[(it * 16 + row) * 9 + 2 * h];
        v2f a0; a0.x = arow[0]; a0.y = arow[1];
        // B fragment (VGPR v, half h -> K = v + 2h)
        v2f b0; b0.x = zs[(2 * h + 0) * 464 + colg];
                b0.y = zs[(2 * h + 1) * 464 + colg];

        v8f c = {0.f, 0.f, 0.f, 0.f, 0.f, 0.f, 0.f, 0.f};
        c = __builtin_amdgcn_wmma_f32_16x16x4_f32(false, a0, false, b0,
                                                  (short)0, c, false, false);
        if (D1 > 4) {  // K = 4..7 second slice (zeros pad beyond D1)
            v2f a1; a1.x = arow[4]; a1.y = arow[5];
            v2f b1; b1.x = zs[(2 * h + 4) * 464 + colg];
                    b1.y = zs[(2 * h + 5) * 464 + colg];
            c = __builtin_amdgcn_wmma_f32_16x16x4_f32(false, a1, false, b1,
                                                      (short)0, c, false, false);
        }
        // C layout: VGPR r, half h -> output row r + 8h
#pragma unroll
        for (int r = 0; r < 8; ++r) {
            int i  = it * 16 + r + 8 * h;
            int ij = i * 32 + j;
            p.out[obase + (long long)ij * D3 + M] = c[r];
        }
    }
}

extern "C" void kernel_launch(void* const* d_in, const int* in_sizes, int n_in,
                              void* d_out, int out_size, void* d_ws, size_t ws_size,
                              hipStream_t stream) {
    (void)in_sizes; (void)out_size; (void)d_ws; (void)ws_size;
    static const int h_l1[NPATHS] = L1_INIT;
    static const int h_l2[NPATHS] = L2_INIT;
    static const int h_l3[NPATHS] = L3_INIT;

    TPParams P;
    for (int l = 0; l < 4; ++l) {
        P.x1[l] = (const float*)d_in[l];
        P.x2[l] = (const float*)d_in[4 + l];
    }
    if (n_in >= 8 + NPATHS) {
        // w3j dict flattened to 44 separate inputs (key order == loop order)
        for (int k = 0; k < NPATHS; ++k) P.w3j[k] = (const float*)d_in[8 + k];
    } else {
        // single concatenated blob fallback
        const float* blob = (const float*)d_in[8];
        size_t off = 0;
        for (int k = 0; k < NPATHS; ++k) {
            P.w3j[k] = blob + off;
            off += (size_t)(2 * h_l1[k] + 1) * (2 * h_l2[k] + 1) * (2 * h_l3[k] + 1);
        }
    }
    P.out = (float*)d_out;

    dim3 grid(NPATHS, 512);
    o3_tp_wmma_kernel<<<grid, 128, 0, stream>>>(P);
}